// GCNLayer_with_22565758173847
// MI455X (gfx1250) — compile-verified
//
#include <hip/hip_runtime.h>

#define N_NODES 100000
#define N_EDGES 1600000
#define D 64

typedef __attribute__((ext_vector_type(2))) float v2f;
typedef __attribute__((ext_vector_type(4))) float v4f;
typedef __attribute__((ext_vector_type(8))) float v8f;

// ---------------------------------------------------------------------------
// Kernel A: h[n][k] = feat[n][k] / out_norm[n]   (float4 vectorized)
// h is staged in d_out (same size as final output; overwritten later).
// ---------------------------------------------------------------------------
__global__ void __launch_bounds__(256) gcn_scale_src(
    const float* __restrict__ feat,
    const float* __restrict__ out_norm,
    float* __restrict__ h) {
  int gid = blockIdx.x * blockDim.x + threadIdx.x;   // one float4 each
  if (gid >= N_NODES * (D / 4)) return;
  int row = gid >> 4;                                // 16 float4 per row
  float inv = 1.0f / out_norm[row];
  v4f v = ((const v4f*)feat)[gid];
  v.x *= inv; v.y *= inv; v.z *= inv; v.w *= inv;
  ((v4f*)h)[gid] = v;
}

// ---------------------------------------------------------------------------
// Kernel B: scatter-add over edges.  One wave32 per edge; each lane moves a
// float2 of the 64-wide row: agg[dst] += h[src].  Both h and agg fit in the
// 192MB L2, so the atomics RMW in L2.
// ---------------------------------------------------------------------------
__global__ void __launch_bounds__(256) gcn_edge_scatter(
    const float* __restrict__ h,
    const int* __restrict__ esrc,
    const int* __restrict__ edst,
    float* __restrict__ agg) {
  unsigned gid = blockIdx.x * blockDim.x + threadIdx.x;
  unsigned e = gid >> 5;
  unsigned lane = gid & 31u;
  if (e >= N_EDGES) return;
  int s = esrc[e];
  int d = edst[e];
  v2f v = ((const v2f*)(h + (size_t)s * D))[lane];
  float* out = agg + (size_t)d * D + 2u * lane;
  unsafeAtomicAdd(out, v.x);        // -> global_atomic_add_f32
  unsafeAtomicAdd(out + 1, v.y);
}

// ---------------------------------------------------------------------------
// Kernel C: out = (agg * (1/in_norm)[:,None]) @ W.T + b   via f32 WMMA.
// One wave per 16-row tile.  V_WMMA_F32_16X16X4_F32 fragment layout (wave32):
//   A (16x4):  lane m = lane%16, khalf = lane/16; VGPR0 = K=2*khalf, VGPR1 = +1
//   B (4x16):  lane n = lane%16;                  VGPR0 = K=2*khalf, VGPR1 = +1
//   C/D:       VGPR j, lanes 0-15 -> M=j, lanes 16-31 -> M=j+8, N = lane%16
// B[k][n] = W[n][k]; bias folded into C init; 1/in_norm folded into A.
// ---------------------------------------------------------------------------
__global__ void __launch_bounds__(256) gcn_wmma_linear(
    const float* __restrict__ agg,
    const float* __restrict__ in_norm,
    const float* __restrict__ W,
    const float* __restrict__ bias,
    float* __restrict__ out) {
  const int lane  = threadIdx.x & 31;
  const int wave  = threadIdx.x >> 5;
  const int tile  = blockIdx.x * 8 + wave;            // 16 rows per tile
  const int NTILES = N_NODES / 16;                    // 6250 (exact)
  if (tile >= NTILES) return;                         // wave-uniform guard

  const int m16   = lane & 15;
  const int khalf = lane >> 4;                        // 0 or 1
  const int mrow  = (tile << 4) + m16;

  const float inv = 1.0f / in_norm[mrow];

  // Hoist all 16 A fragments (K = 0..63 in steps of 4), scaled by 1/in_norm.
  v2f afrag[16];
  const float* arow = agg + (size_t)mrow * D + 2 * khalf;
#pragma unroll
  for (int k = 0; k < 16; ++k) {
    v2f a = *(const v2f*)(arow + 4 * k);
    a.x *= inv; a.y *= inv;
    afrag[k] = a;
  }

#pragma unroll
  for (int nt = 0; nt < 4; ++nt) {
    const int nbase = nt << 4;
    const float bv  = bias[nbase + m16];
    v8f acc = {bv, bv, bv, bv, bv, bv, bv, bv};

    const float* wrow = W + (size_t)(nbase + m16) * D + 2 * khalf;
#pragma unroll
    for (int k = 0; k < 16; ++k) {
      v2f bf = *(const v2f*)(wrow + 4 * k);
      acc = __builtin_amdgcn_wmma_f32_16x16x4_f32(
          /*neg_a=*/false, afrag[k], /*neg_b=*/false, bf,
          /*c_mod=*/(short)0, acc, /*reuse_a=*/false, /*reuse_b=*/false);
    }

    // D write-back: row = tile*16 + j + 8*khalf, col = nbase + m16
    float* orow = out + ((size_t)(tile * 16 + 8 * khalf)) * D + nbase + m16;
#pragma unroll
    for (int j = 0; j < 8; ++j) {
      orow[(size_t)j * D] = acc[j];
    }
  }
}

// ---------------------------------------------------------------------------
extern "C" void kernel_launch(void* const* d_in, const int* in_sizes, int n_in,
                              void* d_out, int out_size, void* d_ws, size_t ws_size,
                              hipStream_t stream) {
  const float* feat     = (const float*)d_in[0];   // [N_NODES, D]
  const float* in_norm  = (const float*)d_in[1];   // [N_NODES]
  const float* out_norm = (const float*)d_in[2];   // [N_NODES]
  const int*   esrc     = (const int*)d_in[3];     // [N_EDGES]
  const int*   edst     = (const int*)d_in[4];     // [N_EDGES]
  const float* W        = (const float*)d_in[5];   // [D, D]
  const float* bias     = (const float*)d_in[6];   // [D]

  float* out = (float*)d_out;
  float* agg = (float*)d_ws;   // N_NODES*D floats = 25.6 MB scratch
  float* h   = out;            // stage h in d_out, overwritten by kernel C

  // zero the accumulation buffer (capturable memset node)
  hipMemsetAsync(agg, 0, (size_t)N_NODES * D * sizeof(float), stream);

  {
    int total = N_NODES * (D / 4);
    gcn_scale_src<<<(total + 255) / 256, 256, 0, stream>>>(feat, out_norm, h);
  }
  {
    long long total = (long long)N_EDGES * 32;
    unsigned blocks = (unsigned)((total + 255) / 256);
    gcn_edge_scatter<<<blocks, 256, 0, stream>>>(h, esrc, edst, agg);
  }
  {
    int ntiles = N_NODES / 16;            // 6250
    int blocks = (ntiles + 7) / 8;        // 8 waves (tiles) per block
    gcn_wmma_linear<<<blocks, 256, 0, stream>>>(agg, in_norm, W, bias, out);
  }
}